// EllipseProposalLayer_78632261255869
// MI455X (gfx1250) — compile-verified
//
#include <hip/hip_runtime.h>
#include <hip/hip_bf16.h>
#include <stdint.h>

typedef unsigned int u32;
typedef unsigned long long u64;
typedef float v2f __attribute__((ext_vector_type(2)));
typedef float v8f __attribute__((ext_vector_type(8)));
typedef u32   v4u __attribute__((ext_vector_type(4)));
typedef u32   v8u __attribute__((ext_vector_type(8)));

#define N_ANCH   327680          // 256*256*5
#define PRE_N    6000
#define POST_N   300
#define NMS_TH   0.7f
#define NWORDS   188             // ceil(6000/32)
#define ROW_T    375             // 6000/16
#define PAD_ROWS 6016            // 188*32
#define HBINS    32768
#define CAP      8192

// ---------------- Tensor Data Mover: 1-D tile load into LDS ----------------
// Builds a D# (group0: 4 SGPRs, group1: 8 SGPRs) for a contiguous run of
// `ndw` dwords and issues TENSOR_LOAD_TO_LDS. Tracked by TENSORcnt.
__device__ __forceinline__ void tdm_load_1d(const void* gptr, u32 lds_off, u32 ndw) {
  u64 ga = (u64)(uintptr_t)gptr;
  v4u g0; v8u g1;
  g0.x = 1u;                                              // count=1 (valid D#)
  g0.y = lds_off;                                         // lds_addr (bytes)
  g0.z = (u32)ga;                                         // global_addr lo
  g0.w = ((u32)(ga >> 32) & 0x01FFFFFFu) | (2u << 30);    // addr hi | type=2
  g1.s0 = (2u << 16);                                     // data_size = 4B
  g1.s1 = (ndw & 0xFFFFu) << 16;                          // tensor_dim0 lo
  g1.s2 = ((ndw >> 16) & 0xFFFFu) | (1u << 16);           // dim0 hi | tensor_dim1=1
  g1.s3 = (ndw & 0xFFFFu) << 16;                          // tile_dim0 = ndw (<=512 here)
  g1.s4 = 0u;                                             // tile_dim1=0, tile_dim2=0
  g1.s5 = ndw;                                            // tensor_dim0_stride lo
  g1.s6 = 0u;
  g1.s7 = 0u;
  asm volatile("tensor_load_to_lds %0, %1" :: "s"(g0), "s"(g1) : "memory");
}

// ---------------- K1: decode + histogram ----------------
__global__ void prep_kernel(const float* __restrict__ cls, const float* __restrict__ ell,
                            const float* __restrict__ anc, float* __restrict__ scores,
                            float* __restrict__ boxes, float* __restrict__ ells,
                            u32* __restrict__ hist) {
  int i = blockIdx.x * 256 + threadIdx.x;
  if (i >= N_ANCH) return;
  float c0 = cls[2*i], c1 = cls[2*i+1];
  float score = 1.f / (1.f + __expf(c0 - c1));            // softmax[...,1]
  float ax1 = anc[4*i], ay1 = anc[4*i+1], ax2 = anc[4*i+2], ay2 = anc[4*i+3];
  float w = ax2 - ax1 + 1.f, h = ay2 - ay1 + 1.f;
  float cx = ax1 + 0.5f*w,   cy = ay1 + 0.5f*h;
  float dx = ell[5*i], dy = ell[5*i+1], dl = ell[5*i+2], ds = ell[5*i+3], dt = ell[5*i+4];
  float px = dx*w + cx, py = dy*h + cy;
  float pl = __expf(dl)*w, ps = __expf(ds)*h, pt = dt;
  float sn, cs; __sincosf(pt, &sn, &cs);
  float hw = 0.5f*sqrtf(pl*cs*pl*cs + ps*sn*ps*sn) + 8.f;
  float hh = 0.5f*sqrtf(pl*sn*pl*sn + ps*cs*ps*cs) + 8.f;
  float x1 = fminf(fmaxf(px - hw, 0.f), 2047.f);
  float y1 = fminf(fmaxf(py - hh, 0.f), 2047.f);
  float x2 = fminf(fmaxf(px + hw, 0.f), 2047.f);
  float y2 = fminf(fmaxf(py + hh, 0.f), 2047.f);
  float bw = x2 - x1 + 1.f, bh = y2 - y1 + 1.f;
  if (!((bw >= 8.f) && (bh >= 8.f))) score = -1.f;
  scores[i] = score;
  boxes[4*i] = x1; boxes[4*i+1] = y1; boxes[4*i+2] = x2; boxes[4*i+3] = y2;
  ells[5*i] = px; ells[5*i+1] = py; ells[5*i+2] = pl; ells[5*i+3] = ps; ells[5*i+4] = pt;
  u32 u = __float_as_uint(score);
  u32 key = (u & 0x80000000u) ? ~u : (u | 0x80000000u);   // order-preserving flip
  atomicAdd(&hist[key >> 17], 1u);
}

__global__ void init_kernel(u32* hist, u32* cnt) {
  int i = blockIdx.x * 256 + threadIdx.x;
  if (i < HBINS) hist[i] = 0u;
  if (i == 0) { cnt[0] = 0u; }
}

// ---------------- K2: radix-select threshold bin ----------------
__global__ void select_kernel(const u32* __restrict__ hist, u32* __restrict__ selB) {
  __shared__ u32 bsum[1024];
  int t = threadIdx.x;
  u32 s = 0;
  for (int b = 0; b < 32; ++b) s += hist[t*32 + b];
  bsum[t] = s;
  __syncthreads();
  if (t == 0) {
    u32 cum = 0; int c = 1023;
    for (; c >= 0; --c) { if (cum + bsum[c] >= (u32)PRE_N) break; cum += bsum[c]; }
    u32 B = 0;
    if (c >= 0) {
      int bin = c*32 + 31;
      for (; bin >= c*32; --bin) { cum += hist[bin]; if (cum >= (u32)PRE_N) break; }
      B = (u32)((bin < c*32) ? c*32 : bin);
    }
    *selB = B;
  }
}

// ---------------- K3: compact candidates ----------------
__global__ void compact_kernel(const float* __restrict__ scores, const u32* __restrict__ selB,
                               u64* __restrict__ cands, u32* __restrict__ cnt) {
  int i = blockIdx.x * 256 + threadIdx.x;
  if (i >= N_ANCH) return;
  u32 u = __float_as_uint(scores[i]);
  u32 key = (u & 0x80000000u) ? ~u : (u | 0x80000000u);
  if ((key >> 17) >= *selB) {
    u32 pos = atomicAdd(cnt, 1u);
    if (pos < CAP) cands[pos] = ((u64)key << 32) | (u64)(0xFFFFFFFFu - (u32)i);
  }
}

// ---------------- K4: in-LDS bitonic sort (descending), emit top 6000 ----------------
__global__ void sort_kernel(const u64* __restrict__ cands, const u32* __restrict__ cnt_p,
                            u32* __restrict__ order, float* __restrict__ tscore) {
  __shared__ u64 sd[CAP];                                  // 64 KB of the 320 KB WGP LDS
  int t = threadIdx.x;
  u32 cnt = *cnt_p; if (cnt > CAP) cnt = CAP;
  for (int i = t; i < CAP; i += 1024) sd[i] = (i < (int)cnt) ? cands[i] : 0ull;
  __syncthreads();
  for (int k = 2; k <= CAP; k <<= 1) {
    for (int j = k >> 1; j > 0; j >>= 1) {
      for (int i = t; i < CAP; i += 1024) {
        int ixj = i ^ j;
        if (ixj > i) {
          bool desc = ((i & k) == 0);
          u64 a = sd[i], b = sd[ixj];
          if (desc ? (a < b) : (a > b)) { sd[i] = b; sd[ixj] = a; }
        }
      }
      __syncthreads();
    }
  }
  for (int i = t; i < PRE_N; i += 1024) {
    u64 v = sd[i];
    u32 key = (u32)(v >> 32);
    u32 ub  = (key & 0x80000000u) ? (key ^ 0x80000000u) : ~key;
    order[i]  = 0xFFFFFFFFu - (u32)(v & 0xFFFFFFFFu);
    tscore[i] = __uint_as_float(ub);
  }
}

// ---------------- K5: gather top-6000 boxes/ellipses (+zero pad rows) ----------------
__global__ void gather_kernel(const float* __restrict__ boxes, const float* __restrict__ ells,
                              const u32* __restrict__ order, float* __restrict__ tbox,
                              float* __restrict__ tell) {
  int i = blockIdx.x * 256 + threadIdx.x;
  if (i < PRE_N) {
    u32 o = order[i]; if (o >= (u32)N_ANCH) o = 0;
    #pragma unroll
    for (int c = 0; c < 4; ++c) tbox[i*4+c] = boxes[o*4+c];
    #pragma unroll
    for (int c = 0; c < 5; ++c) tell[i*5+c] = ells[o*5+c];
  } else if (i < PAD_ROWS) {
    #pragma unroll
    for (int c = 0; c < 4; ++c) tbox[i*4+c] = 0.f;
  }
}

// ---------------- K6: NMS suppression mask (TDM staging + WMMA area-sums) ----------------
__device__ __forceinline__ float box_area(const float* b) {
  return (b[2] - b[0] + 1.f) * (b[3] - b[1] + 1.f);
}

__global__ void nms_mask_kernel(const float* __restrict__ tbox, u32* __restrict__ mask) {
  __shared__ float colb[32*4];   // 32 column boxes
  __shared__ float rowb[128*4];  // 128 row boxes (16 per wave * 8 waves)
  const int cw = blockIdx.x;     // column word 0..187 (32 cols)
  const int by = blockIdx.y;     // row super-tile 0..46 (128 rows)
  if (threadIdx.x == 0) {        // one wave issues the TDM DMAs (EXEC ignored)
    tdm_load_1d(tbox + cw*128, (u32)(uintptr_t)colb, 128);
    tdm_load_1d(tbox + by*512, (u32)(uintptr_t)rowb, 512);
    __builtin_amdgcn_s_wait_tensorcnt(0);
  }
  __syncthreads();
  const int wave = threadIdx.x >> 5, lane = threadIdx.x & 31;
  const int rt = by*8 + wave;
  if (rt >= ROW_T) return;

  // A = [area_row, 1] (16x4, K2..3 zero), B_h = [1; area_col_h] (4x16).
  // D[k] @ lane l == area(row k+8*(l>=16)) + area(col l%16): exactly the
  // per-lane IoU denominator term for the 16x16 tile.
  v2f a; a.x = (lane < 16) ? box_area(&rowb[(wave*16 + lane)*4]) : 0.f;
         a.y = (lane < 16) ? 1.f : 0.f;
  v2f b0; b0.x = (lane < 16) ? 1.f : 0.f;
          b0.y = (lane < 16) ? box_area(&colb[lane*4]) : 0.f;
  v2f b1; b1.x = (lane < 16) ? 1.f : 0.f;
          b1.y = (lane < 16) ? box_area(&colb[(16 + lane)*4]) : 0.f;
  v8f cz = {};
  v8f s0 = __builtin_amdgcn_wmma_f32_16x16x4_f32(false, a, false, b0, (short)0, cz, false, false);
  v8f s1 = __builtin_amdgcn_wmma_f32_16x16x4_f32(false, a, false, b1, (short)0, cz, false, false);

  const int rlocal_base = wave*16 + ((lane & 16) ? 8 : 0);
  const int cl = lane & 15;
  #pragma unroll
  for (int k = 0; k < 8; ++k) {
    const int rlocal = rlocal_base + k;
    const float* rb = &rowb[rlocal*4];
    const int row = by*128 + rlocal;
    u32 bal[2];
    #pragma unroll
    for (int h = 0; h < 2; ++h) {
      const float* cb = &colb[(h*16 + cl)*4];
      const int col = cw*32 + h*16 + cl;
      float iw = fminf(rb[2], cb[2]) - fmaxf(rb[0], cb[0]) + 1.f;
      float ih = fminf(rb[3], cb[3]) - fmaxf(rb[1], cb[1]) + 1.f;
      float inter = fmaxf(iw, 0.f) * fmaxf(ih, 0.f);
      float sa = (h == 0) ? s0[k] : s1[k];
      bool pred = (inter > NMS_TH * (sa - inter)) && (col > row) && (col < PRE_N);
      bal[h] = __builtin_amdgcn_ballot_w32(pred);
    }
    if (lane == 0) {
      const int r0 = by*128 + wave*16 + k;                  // rows k and k+8
      mask[r0*NWORDS + cw]       = (bal[0] & 0xFFFFu) | ((bal[1] & 0xFFFFu) << 16);
      mask[(r0 + 8)*NWORDS + cw] = (bal[0] >> 16)     |  (bal[1] & 0xFFFF0000u);
    }
  }
}

// ---------------- K7: sequential greedy keep + first-300 scan ----------------
__global__ void nms_seq_kernel(const float* __restrict__ tscore, const u32* __restrict__ mask,
                               u32* __restrict__ kept, u32* __restrict__ nkeep) {
  __shared__ u32 keep[NWORDS];
  int t = threadIdx.x;
  if (t < NWORDS) {
    u32 w = 0;
    for (int b = 0; b < 32; ++b) {
      int j = t*32 + b;
      if (j < PRE_N && tscore[j] > -0.5f) w |= (1u << b);
    }
    keep[t] = w;
  }
  for (int i = 0; i < PRE_N; ++i) {
    __syncthreads();
    u32 wi = keep[i >> 5];
    __syncthreads();
    if ((wi >> (i & 31)) & 1u) {
      if (t < NWORDS) keep[t] &= ~mask[i*NWORDS + t];
    }
  }
  __syncthreads();
  if (t == 0) {
    u32 nk = 0;
    for (int i = 0; i < PRE_N && nk < POST_N; ++i)
      if ((keep[i >> 5] >> (i & 31)) & 1u) kept[nk++] = (u32)i;
    for (u32 j = nk; j < POST_N; ++j) kept[j] = PRE_N - 1;  // fill_value
    *nkeep = nk;
  }
}

// ---------------- K8: write outputs ----------------
__global__ void write_out_kernel(const float* __restrict__ tbox, const float* __restrict__ tell,
                                 const float* __restrict__ tscore, const u32* __restrict__ kept,
                                 const u32* __restrict__ nkeep_p, float* __restrict__ out) {
  int i = blockIdx.x * 256 + threadIdx.x;
  if (i >= POST_N) return;
  bool v = (u32)i < *nkeep_p;
  u32 k = kept[i];
  #pragma unroll
  for (int c = 0; c < 4; ++c) out[i*4 + c] = v ? tbox[k*4 + c] : 0.f;
  #pragma unroll
  for (int c = 0; c < 5; ++c) out[POST_N*4 + i*5 + c] = v ? tell[k*5 + c] : 0.f;
  out[POST_N*9 + i] = v ? tscore[k] : 0.f;
}

// ---------------- workspace layout (bytes) ----------------
#define OFF_SCORES  0u
#define OFF_BOXES   1310720u
#define OFF_ELL     6553600u
#define OFF_HIST    13107200u
#define OFF_CNT     13238272u
#define OFF_CANDS   13238400u
#define OFF_ORDER   13303936u
#define OFF_TSCORE  13327936u
#define OFF_TBOX    13351936u
#define OFF_TELL    13448192u
#define OFF_MASK    13568192u
#define OFF_KEPT    18080192u
#define OFF_NKEEP   18081408u

extern "C" void kernel_launch(void* const* d_in, const int* in_sizes, int n_in,
                              void* d_out, int out_size, void* d_ws, size_t ws_size,
                              hipStream_t stream) {
  const float* cls = (const float*)d_in[0];
  const float* ell = (const float*)d_in[1];
  const float* anc = (const float*)d_in[2];
  char* ws = (char*)d_ws;
  float* scores = (float*)(ws + OFF_SCORES);
  float* boxes  = (float*)(ws + OFF_BOXES);
  float* ells   = (float*)(ws + OFF_ELL);
  u32*   hist   = (u32*)  (ws + OFF_HIST);
  u32*   cnt    = (u32*)  (ws + OFF_CNT);      // [0]=cand count
  u32*   selB   = cnt + 2;
  u64*   cands  = (u64*)  (ws + OFF_CANDS);
  u32*   order  = (u32*)  (ws + OFF_ORDER);
  float* tscore = (float*)(ws + OFF_TSCORE);
  float* tbox   = (float*)(ws + OFF_TBOX);
  float* tell   = (float*)(ws + OFF_TELL);
  u32*   mask   = (u32*)  (ws + OFF_MASK);
  u32*   kept   = (u32*)  (ws + OFF_KEPT);
  u32*   nkeep  = (u32*)  (ws + OFF_NKEEP);
  float* out    = (float*)d_out;

  init_kernel<<<HBINS/256, 256, 0, stream>>>(hist, cnt);
  prep_kernel<<<(N_ANCH + 255)/256, 256, 0, stream>>>(cls, ell, anc, scores, boxes, ells, hist);
  select_kernel<<<1, 1024, 0, stream>>>(hist, selB);
  compact_kernel<<<(N_ANCH + 255)/256, 256, 0, stream>>>(scores, selB, cands, cnt);
  sort_kernel<<<1, 1024, 0, stream>>>(cands, cnt, order, tscore);
  gather_kernel<<<(PAD_ROWS + 255)/256, 256, 0, stream>>>(boxes, ells, order, tbox, tell);
  nms_mask_kernel<<<dim3(NWORDS, 47), 256, 0, stream>>>(tbox, mask);
  nms_seq_kernel<<<1, 256, 0, stream>>>(tscore, mask, kept, nkeep);
  write_out_kernel<<<2, 256, 0, stream>>>(tbox, tell, tscore, kept, nkeep, out);
}